// GNNEncoder_592705487106
// MI455X (gfx1250) — compile-verified
//
#include <hip/hip_runtime.h>

typedef float v2f __attribute__((ext_vector_type(2)));
typedef float v8f __attribute__((ext_vector_type(8)));

#define N_PAPER   100000
#define N_AUTHOR  40000
#define F_PAPER   128
#define F_AUTHOR  64
#define HIDF      128
#define OUTF      128
#define E_CITES   800000
#define E_WRITES  400000

// ---------------------------------------------------------------------------
// Zero a contiguous float region (grid-stride).
__global__ void zero_kernel(float* __restrict__ p, long long n) {
    long long i = (long long)blockIdx.x * blockDim.x + threadIdx.x;
    long long s = (long long)gridDim.x * blockDim.x;
    for (; i < n; i += s) p[i] = 0.0f;
}

// ---------------------------------------------------------------------------
// Degree count: one thread per edge, fp32 atomic (reference uses float counts).
__global__ void count_kernel(const int* __restrict__ dst, float* __restrict__ cnt, int E) {
    int e = blockIdx.x * blockDim.x + threadIdx.x;
    if (e < E) atomicAdd(&cnt[dst[e]], 1.0f);
}

// cnt -> 1/max(cnt,1)  (in place)
__global__ void rinv_kernel(float* __restrict__ c, int n) {
    int i = blockIdx.x * blockDim.x + threadIdx.x;
    if (i < n) c[i] = 1.0f / fmaxf(c[i], 1.0f);
}

// ---------------------------------------------------------------------------
// Edge scatter: 32 lanes per edge, float4 per lane (F = 128 fixed).
// sum[dst] += x[src]
__global__ void scatter_kernel(const float* __restrict__ x,
                               const int* __restrict__ src,
                               const int* __restrict__ dst,
                               float* __restrict__ sum, int E) {
    int t = blockIdx.x * blockDim.x + threadIdx.x;
    int e = t >> 5;
    if (e >= E) return;
    int c = (t & 31) << 2;                       // feature chunk 0..124
    const float4 v = *reinterpret_cast<const float4*>(x + (size_t)src[e] * 128 + c);
    float* o = sum + (size_t)dst[e] * 128 + c;
    atomicAdd(o + 0, v.x);
    atomicAdd(o + 1, v.y);
    atomicAdd(o + 2, v.z);
    atomicAdd(o + 3, v.w);
}

// ---------------------------------------------------------------------------
// Fused SAGE output GEMM using V_WMMA_F32_16X16X4_F32 (exact fp32 path):
//   Out[m,n] = (Asum[m,:] * rinv[m]) . Wl[n,:]  +  Xdst[m,:] . Wr[n,:]  + bl[n]
// One wave per 16x16 output tile; 4 waves (128 threads) per block.
// A-frag (16x4 f32): lanes 0-15 hold K={k,k+1}, lanes 16-31 hold K={k+2,k+3}.
// B-frag (4x16 f32): mirrored, per-lane row of Wl (row-major [128,K]) gives B^T.
// C/D (16x16 f32, 8 VGPRs): vgpr r, lanes 0-15 -> M=r, lanes 16-31 -> M=r+8.
__global__ __launch_bounds__(128)
void sage_gemm_kernel(const float* __restrict__ Asum, const float* __restrict__ rinv,
                      const float* __restrict__ Xdst, const float* __restrict__ Wl,
                      const float* __restrict__ bl,   const float* __restrict__ Wr,
                      float* __restrict__ Out, int M, int K, int Kr, int do_relu) {
    const int wave = threadIdx.x >> 5;
    const int lane = threadIdx.x & 31;
    const int tile = blockIdx.x * 4 + wave;
    const int tilesN = OUTF / 16;                 // 8
    const int numTiles = (M >> 4) * tilesN;
    if (tile >= numTiles) return;                 // wave-uniform: EXEC stays all-ones

    const int tm   = tile / tilesN;
    const int tn   = tile % tilesN;
    const int half = lane >> 4;                   // 0: K lo pair, 1: K hi pair
    const int l16  = lane & 15;
    const int rowA = tm * 16 + l16;               // A row for this lane
    const int colB = tn * 16 + l16;               // output column / Wl,Wr row

    const float rv = rinv[rowA];

    const float* ap = Asum + (size_t)rowA * K + half * 2;
    const float* bp = Wl   + (size_t)colB * K + half * 2;

    v8f acc = {};
    #pragma unroll 4
    for (int k = 0; k < K; k += 4) {
        v2f a = *(const v2f*)(ap + k);
        a *= rv;                                  // fold mean normalization into A
        v2f b = *(const v2f*)(bp + k);
        acc = __builtin_amdgcn_wmma_f32_16x16x4_f32(
                  false, a, false, b, (short)0, acc, false, false);
    }

    const float* ap2 = Xdst + (size_t)rowA * Kr + half * 2;
    const float* bp2 = Wr   + (size_t)colB * Kr + half * 2;
    #pragma unroll 4
    for (int k = 0; k < Kr; k += 4) {
        v2f a = *(const v2f*)(ap2 + k);
        v2f b = *(const v2f*)(bp2 + k);
        acc = __builtin_amdgcn_wmma_f32_16x16x4_f32(
                  false, a, false, b, (short)0, acc, false, false);
    }

    const float bias = bl[colB];
    float* orow = Out + (size_t)(tm * 16 + half * 8) * OUTF + colB;
    #pragma unroll
    for (int r = 0; r < 8; ++r) {
        float v = acc[r] + bias;
        if (do_relu) v = fmaxf(v, 0.0f);
        orow[(size_t)r * OUTF] = v;
    }
}

// ---------------------------------------------------------------------------
extern "C" void kernel_launch(void* const* d_in, const int* in_sizes, int n_in,
                              void* d_out, int out_size, void* d_ws, size_t ws_size,
                              hipStream_t stream) {
    (void)in_sizes; (void)n_in; (void)out_size; (void)ws_size;

    const float* x_paper  = (const float*)d_in[0];
    const float* x_author = (const float*)d_in[1];
    const int*   c_src    = (const int*)d_in[2];
    const int*   c_dst    = (const int*)d_in[3];
    const int*   w_src    = (const int*)d_in[4];
    const int*   w_dst    = (const int*)d_in[5];
    const float* Wl1c = (const float*)d_in[6];
    const float* bl1c = (const float*)d_in[7];
    const float* Wr1c = (const float*)d_in[8];
    const float* Wl1w = (const float*)d_in[9];
    const float* bl1w = (const float*)d_in[10];
    const float* Wr1w = (const float*)d_in[11];
    const float* Wl2c = (const float*)d_in[12];
    const float* bl2c = (const float*)d_in[13];
    const float* Wr2c = (const float*)d_in[14];
    const float* Wl2w = (const float*)d_in[15];
    const float* bl2w = (const float*)d_in[16];
    const float* Wr2w = (const float*)d_in[17];

    // workspace layout (floats)
    float* ws       = (float*)d_ws;
    float* sum_c    = ws;                                     // N_PAPER*128
    float* sum_w    = sum_c + (size_t)N_PAPER * 128;          // N_AUTHOR*128
    float* rinv_c   = sum_w + (size_t)N_AUTHOR * 128;         // N_PAPER
    float* rinv_w   = rinv_c + N_PAPER;                       // N_AUTHOR
    float* h_paper  = rinv_w + N_AUTHOR;                      // N_PAPER*128
    float* h_author = h_paper + (size_t)N_PAPER * 128;        // N_AUTHOR*128

    float* o_paper  = (float*)d_out;                          // N_PAPER*128
    float* o_author = o_paper + (size_t)N_PAPER * 128;        // N_AUTHOR*128

    const long long sums_n = (long long)N_PAPER * 128 + (long long)N_AUTHOR * 128;
    const long long cnts_n = N_PAPER + N_AUTHOR;

    // 1. zero sums + counts (contiguous region)
    zero_kernel<<<4096, 256, 0, stream>>>(sum_c, sums_n + cnts_n);

    // 2. degrees (same for both layers) -> rinv
    count_kernel<<<(E_CITES + 255) / 256, 256, 0, stream>>>(c_dst, rinv_c, E_CITES);
    count_kernel<<<(E_WRITES + 255) / 256, 256, 0, stream>>>(w_dst, rinv_w, E_WRITES);

    // 3. layer-1 scatter (messages are x_paper rows for both relations)
    scatter_kernel<<<E_CITES / 8, 256, 0, stream>>>(x_paper, c_src, c_dst, sum_c, E_CITES);
    scatter_kernel<<<E_WRITES / 8, 256, 0, stream>>>(x_paper, w_src, w_dst, sum_w, E_WRITES);

    // 4. rinv = 1/max(cnt,1)
    rinv_kernel<<<(N_PAPER + 255) / 256, 256, 0, stream>>>(rinv_c, N_PAPER);
    rinv_kernel<<<(N_AUTHOR + 255) / 256, 256, 0, stream>>>(rinv_w, N_AUTHOR);

    // 5. layer-1 GEMMs (ReLU)
    sage_gemm_kernel<<<(N_PAPER / 16) * 8 / 4, 128, 0, stream>>>(
        sum_c, rinv_c, x_paper, Wl1c, bl1c, Wr1c, h_paper,
        N_PAPER, F_PAPER, F_PAPER, 1);
    sage_gemm_kernel<<<(N_AUTHOR / 16) * 8 / 4, 128, 0, stream>>>(
        sum_w, rinv_w, x_author, Wl1w, bl1w, Wr1w, h_author,
        N_AUTHOR, F_PAPER, F_AUTHOR, 1);

    // 6. layer-2: re-zero sums, scatter h_paper
    zero_kernel<<<4096, 256, 0, stream>>>(sum_c, sums_n);
    scatter_kernel<<<E_CITES / 8, 256, 0, stream>>>(h_paper, c_src, c_dst, sum_c, E_CITES);
    scatter_kernel<<<E_WRITES / 8, 256, 0, stream>>>(h_paper, w_src, w_dst, sum_w, E_WRITES);

    // 7. layer-2 GEMMs (no ReLU) -> d_out
    sage_gemm_kernel<<<(N_PAPER / 16) * 8 / 4, 128, 0, stream>>>(
        sum_c, rinv_c, h_paper, Wl2c, bl2c, Wr2c, o_paper,
        N_PAPER, HIDF, HIDF, 0);
    sage_gemm_kernel<<<(N_AUTHOR / 16) * 8 / 4, 128, 0, stream>>>(
        sum_w, rinv_w, h_author, Wl2w, bl2w, Wr2w, o_author,
        N_AUTHOR, HIDF, HIDF, 0);
}